// InterpolatingBSpline1d_8735963480088
// MI455X (gfx1250) — compile-verified
//
#include <hip/hip_runtime.h>

typedef float v2f __attribute__((ext_vector_type(2)));
typedef float v8f __attribute__((ext_vector_type(8)));
typedef float f4  __attribute__((ext_vector_type(4)));

#define MSZ 1024
#define NSZ 4194304
#define TILES_PER_WAVE 16
#define WAVES_PER_BLOCK 8

// ---------------------------------------------------------------------------
// Kernel 1: exact-elimination spline solve.
// Full (M+2)^2 system reduces exactly (Gaussian elimination of boundary rows) to:
//   c1 = d0 ; cM = d[M-1] ; interior i=2..M-1: (1/6)c[i-1]+(2/3)c[i]+(1/6)c[i+1]=d[i-1]
//   c0 = 2c1 - c2 ; c[M+1] = 2cM - c[M-1]
// Constant-coefficient Thomas per channel (8 threads). Writes coefsT[pos][ch].
// ---------------------------------------------------------------------------
__global__ __launch_bounds__(32) void spline_solve_kernel(const float* __restrict__ data,
                                                          float* __restrict__ coefsT) {
  __shared__ float gbuf[8][1024];
  __shared__ float wbuf[8][1024];
  const int ch = threadIdx.x;
  if (ch < 8) {
    const float* d = data + ch * MSZ;
    const float a = 1.0f / 6.0f, b = 2.0f / 3.0f;
    const float c1 = d[0];
    const float cM = d[MSZ - 1];
    float w = b;
    float invw = __builtin_amdgcn_rcpf(w);
    float g = d[1] - a * c1;                       // r_2
    gbuf[ch][2] = g; wbuf[ch][2] = invw;
    for (int i = 3; i <= MSZ - 1; ++i) {
      float r = (i == MSZ - 1) ? (d[MSZ - 2] - a * cM) : d[i - 1];
      float m = a * invw;
      w = b - m * a;
      invw = __builtin_amdgcn_rcpf(w);
      g = r - m * g;
      gbuf[ch][i] = g; wbuf[ch][i] = invw;
    }
    float xlast = g * invw;                        // x_{M-1}
    coefsT[(MSZ - 1) * 8 + ch] = xlast;
    float xn = xlast;
    for (int i = MSZ - 2; i >= 2; --i) {
      float xi = (gbuf[ch][i] - a * xn) * wbuf[ch][i];
      coefsT[i * 8 + ch] = xi;
      xn = xi;
    }
    coefsT[1 * 8 + ch]         = c1;
    coefsT[MSZ * 8 + ch]       = cM;
    coefsT[0 * 8 + ch]         = 2.0f * c1 - xn;      // xn == x_2
    coefsT[(MSZ + 1) * 8 + ch] = 2.0f * cM - xlast;   // 2*cM - c[M-1]
  }
}

// ---------------------------------------------------------------------------
// Kernel 2: streaming evaluation. 16 samples per wave-iteration; lanes s and
// s+16 handle channels 0-3 / 4-7 of sample s. Basis weights come from one
// v_wmma_f32_16x16x4_f32: A = BASIS (rows 0-3 and 8-11), B columns = t-powers.
// Coef table staged into LDS with CDNA5 async-to-LDS loads.
// ---------------------------------------------------------------------------
__global__ __launch_bounds__(256) void spline_eval_kernel(const float* __restrict__ u,
                                                          const float* __restrict__ coefsT,
                                                          float* __restrict__ out) {
  __shared__ float lds_cof[(MSZ + 2) * 8];          // 32832 B

  // --- async-stage coefsT (global -> LDS), tracked by ASYNCcnt ---
  const int NCHUNK = ((MSZ + 2) * 8) / 4;           // 2052 x b128
  for (int k = threadIdx.x; k < NCHUNK; k += blockDim.x) {
    unsigned ldsoff = (unsigned)(unsigned long long)(&lds_cof[k * 4]);
    unsigned long long gaddr = (unsigned long long)(coefsT + k * 4);
    asm volatile("global_load_async_to_lds_b128 %0, %1, off"
                 :: "v"(ldsoff), "v"(gaddr) : "memory");
  }
  asm volatile("s_wait_asynccnt 0x0" ::: "memory");
  __syncthreads();

  const int lane = threadIdx.x & 31;
  const int row  = lane & 15;
  const bool hi  = lane >= 16;

  // A-matrix: per-lane constants. Lane L (<16) holds A[L,0],A[L,1]; lane L+16
  // holds A[L,2],A[L,3]. Rows 0-3 and 8-11 = BASIS/6, rows 4-7,12-15 = 0.
  const int  br  = row & 3;
  const bool act = (row & 4) == 0;
  const float k6 = 1.0f / 6.0f;
  float a0, a1;
  if (!hi) {
    a0 = (br == 0 ? -1.0f : br == 1 ?  3.0f : br == 2 ? -3.0f : 1.0f) * k6;
    a1 = (br == 0 ?  3.0f : br == 1 ? -6.0f : br == 2 ?  3.0f : 0.0f) * k6;
  } else {
    a0 = (br == 0 ? -3.0f : br == 1 ?  0.0f : br == 2 ?  3.0f : 0.0f) * k6;
    a1 = (br == 0 ?  1.0f : br == 1 ?  4.0f : br == 2 ?  1.0f : 0.0f) * k6;
  }
  if (!act) { a0 = 0.0f; a1 = 0.0f; }
  const v2f Am = {a0, a1};

  const int waveg = blockIdx.x * WAVES_PER_BLOCK + (threadIdx.x >> 5);
  const int choff = hi ? 4 : 0;

  for (int it = 0; it < TILES_PER_WAVE; ++it) {
    const int tile = waveg * TILES_PER_WAVE + it;
    const int s = tile * 16 + row;

    float un = u[s] * (float)(MSZ - 1);
    float fi = floorf(un);
    int   i  = (int)fi;
    float t  = un - fi;
    if (un < 0.0f)              { i = 0;       t = un; }
    if (un >= (float)(MSZ - 2)) { i = MSZ - 2; t = un - (float)(MSZ - 2); }
    float t2 = t * t;
    float t3 = t2 * t;

    // B column s = [t^3, t^2, t, 1]; low-K half in lanes 0-15, high-K in 16-31.
    v2f Bm = { hi ? t : t3, hi ? 1.0f : t2 };
    v8f Cm = {0.0f, 0.0f, 0.0f, 0.0f, 0.0f, 0.0f, 0.0f, 0.0f};
    v8f Dm = __builtin_amdgcn_wmma_f32_16x16x4_f32(false, Am, false, Bm,
                                                   (short)0, Cm, false, false);
    // Lane s: Dm[f]=D[f,s]=bv[s][f]; lane s+16: Dm[f]=D[8+f,s]=bv[s][f].
    f4 acc = {0.0f, 0.0f, 0.0f, 0.0f};
#pragma unroll
    for (int f = 0; f < 4; ++f) {
      const f4 cp = *(const f4*)&lds_cof[(i + f) * 8 + choff];
      acc += Dm[f] * cp;
    }
    __builtin_nontemporal_store(acc, (f4*)(out + (size_t)s * 8 + choff));
  }
}

extern "C" void kernel_launch(void* const* d_in, const int* in_sizes, int n_in,
                              void* d_out, int out_size, void* d_ws, size_t ws_size,
                              hipStream_t stream) {
  (void)in_sizes; (void)n_in; (void)out_size; (void)ws_size;
  const float* u    = (const float*)d_in[0];   // (N,1) coords
  const float* data = (const float*)d_in[1];   // (C,M) grid values
  float* out    = (float*)d_out;               // (N,C)
  float* coefsT = (float*)d_ws;                // (M+2) x 8 transposed coefs

  spline_solve_kernel<<<1, 32, 0, stream>>>(data, coefsT);

  const int blocks = NSZ / (16 * TILES_PER_WAVE * WAVES_PER_BLOCK); // 2048
  spline_eval_kernel<<<blocks, 256, 0, stream>>>(u, coefsT, out);
}